// PositionWisedTableEncoder_51522427683109
// MI455X (gfx1250) — compile-verified
//
#include <hip/hip_runtime.h>
#include <hip/hip_bf16.h>

typedef __attribute__((ext_vector_type(2))) float v2f;
typedef __attribute__((ext_vector_type(4))) float v4f;
typedef __attribute__((ext_vector_type(8))) float v8f;

#define HDIM  768
#define SEQ   128
#define BATCH 4
#define ROWS  (BATCH * SEQ)   // 512 rows per tensor
#define LDW   (2 * HDIM)      // W row stride = 1536

// ---------------------------------------------------------------------------
// Phase 1: rotary embedding for both tensors into workspace.
// grid = 1024 blocks (512 review rows, then 512 reply rows), block = 384
// threads, one (even,odd) pair per thread. Inputs are read exactly once ->
// non-temporal loads keep them from polluting WGP$/L2.
// ---------------------------------------------------------------------------
__global__ void rotary_kernel(const float* __restrict__ review,
                              const float* __restrict__ reply,
                              float* __restrict__ arev,
                              float* __restrict__ arep) {
    int row = blockIdx.x;            // 0..1023
    int i   = threadIdx.x;           // pair index 0..383
    const float* src;
    float* dst;
    if (row < ROWS) { src = review + row * HDIM;          dst = arev + row * HDIM; }
    else            { src = reply  + (row - ROWS) * HDIM; dst = arep + (row - ROWS) * HDIM; }
    int s = row & (SEQ - 1);         // position within sequence
    float inv = __powf(10000.0f, -2.0f * (float)i / (float)HDIM);
    float ang = (float)s * inv;
    float sn, cs;
    __sincosf(ang, &sn, &cs);
    v2f x = __builtin_nontemporal_load((const v2f*)(src + 2 * i));
    v2f r;
    r.x = x.x * cs - x.y * sn;       // x*cos + (-x_odd)*sin
    r.y = x.y * cs + x.x * sn;       // x*cos + ( x_even)*sin
    *(v2f*)(dst + 2 * i) = r;        // re-read by GEMM soon -> regular temporal
}

// ---------------------------------------------------------------------------
// Phase 2: P = A @ Wslice^T via V_WMMA_F32_16X16X4_F32 (full fp32 precision).
// One wave computes a 16x64 strip of P (4 accumulators reuse each A fragment).
// Wave ids: [0,384) -> review/W[:, :H], [384,768) -> reply/W[:, H:].
// A fragment (16x4): lane m = lane%16, K-pair offset = (lane>=16)*2 -> float2.
// B fragment (4x16): lane j = lane%16 selects W row o0+j; same K-pair -> float2.
// D (16x16 f32): VGPR q -> rows q (lanes 0-15) and q+8 (lanes 16-31), col=lane%16.
// All operands are L2-resident (A's 3 MB + W 4.5 MB << 192 MB L2).
// ---------------------------------------------------------------------------
__global__ void gemm_wmma_kernel(const float* __restrict__ arev,
                                 const float* __restrict__ arep,
                                 const float* __restrict__ W,
                                 float* __restrict__ prev,
                                 float* __restrict__ prep) {
    int wid  = (blockIdx.x * blockDim.x + threadIdx.x) >> 5;  // 0..767
    int lane = threadIdx.x & 31;

    int g  = wid / 384;              // 0 = review path, 1 = reply path
    int r  = wid % 384;
    int n0 = (r / 12) * 16;          // row tile base (0..496)
    int o0 = (r % 12) * 64;          // output-col strip base (0..704)

    const float* A  = g ? arep : arev;
    const float* Wg = W + (g ? HDIM : 0);
    float*       P  = g ? prep : prev;

    int mr = lane & 15;              // matrix row (A) / W row within tile (B)
    int kh = (lane >> 4) << 1;       // K half: 0 or 2

    const float* Arow = A + (n0 + mr) * HDIM + kh;
    const float* W0   = Wg + (o0 +  0 + mr) * LDW + kh;
    const float* W1   = Wg + (o0 + 16 + mr) * LDW + kh;
    const float* W2   = Wg + (o0 + 32 + mr) * LDW + kh;
    const float* W3   = Wg + (o0 + 48 + mr) * LDW + kh;

    v8f c0 = {}, c1 = {}, c2 = {}, c3 = {};
    for (int k = 0; k < HDIM; k += 4) {
        v2f a  = *(const v2f*)(Arow + k);
        v2f b0 = *(const v2f*)(W0 + k);
        v2f b1 = *(const v2f*)(W1 + k);
        v2f b2 = *(const v2f*)(W2 + k);
        v2f b3 = *(const v2f*)(W3 + k);
        c0 = __builtin_amdgcn_wmma_f32_16x16x4_f32(false, a, false, b0, (short)0, c0, false, false);
        c1 = __builtin_amdgcn_wmma_f32_16x16x4_f32(false, a, false, b1, (short)0, c1, false, false);
        c2 = __builtin_amdgcn_wmma_f32_16x16x4_f32(false, a, false, b2, (short)0, c2, false, false);
        c3 = __builtin_amdgcn_wmma_f32_16x16x4_f32(false, a, false, b3, (short)0, c3, false, false);
    }

    int orow = n0 + (lane >> 4) * 8;
    int ocol = o0 + (lane & 15);
    float* Pp = P + orow * HDIM + ocol;
#pragma unroll
    for (int q = 0; q < 8; ++q) {
        Pp[q * HDIM +  0] = c0[q];   // re-read 128x by phase 3 -> keep temporal
        Pp[q * HDIM + 16] = c1[q];
        Pp[q * HDIM + 32] = c2[q];
        Pp[q * HDIM + 48] = c3[q];
    }
}

// ---------------------------------------------------------------------------
// Phase 3: out[b,n,m,:] = relu(prev[b,n,:] + prep[b,m,:] + bias) -- the 201 MB
// streaming-store phase that sets the kernel's roofline (~8.6 us @ 23.3 TB/s).
// One 768-float row per block, float4 per lane. Output is write-once ->
// non-temporal global_store_b128 so the stream doesn't thrash L2 and evict the
// hot 3 MB of projections that every block re-reads.
// ---------------------------------------------------------------------------
__global__ void bcast_relu_kernel(const float* __restrict__ prev,
                                  const float* __restrict__ prep,
                                  const float* __restrict__ bias,
                                  float* __restrict__ out) {
    int rowid = blockIdx.x;          // ((b*128 + n)*128 + m)
    int bn = rowid >> 7;             // b*128 + n
    int m  = rowid & 127;
    int b  = bn >> 7;

    const v4f* pr = (const v4f*)(prev + bn * HDIM);
    const v4f* pp = (const v4f*)(prep + (b * SEQ + m) * HDIM);
    const v4f* bb = (const v4f*)bias;
    v4f* o = (v4f*)out + (size_t)rowid * (HDIM / 4);

    int t = threadIdx.x;             // 0..191
    v4f v = pr[t] + pp[t] + bb[t];
    v.x = fmaxf(v.x, 0.0f);
    v.y = fmaxf(v.y, 0.0f);
    v.z = fmaxf(v.z, 0.0f);
    v.w = fmaxf(v.w, 0.0f);
    __builtin_nontemporal_store(v, &o[t]);
}

extern "C" void kernel_launch(void* const* d_in, const int* in_sizes, int n_in,
                              void* d_out, int out_size, void* d_ws, size_t ws_size,
                              hipStream_t stream) {
    (void)in_sizes; (void)n_in; (void)out_size; (void)ws_size;

    const float* review = (const float*)d_in[0];   // (4,128,768)
    const float* reply  = (const float*)d_in[1];   // (4,128,768)
    const float* W      = (const float*)d_in[2];   // (768,1536) row-major
    const float* bias   = (const float*)d_in[3];   // (768,)

    float* arev = (float*)d_ws;              // 512*768 rotated review
    float* arep = arev + ROWS * HDIM;        // 512*768 rotated reply
    float* prev = arep + ROWS * HDIM;        // 512*768 proj_review
    float* prep = prev + ROWS * HDIM;        // 512*768 proj_reply
    // total workspace: 4 * 1.5 MB = 6 MB

    rotary_kernel<<<2 * ROWS, HDIM / 2, 0, stream>>>(review, reply, arev, arep);
    gemm_wmma_kernel<<<96, 256, 0, stream>>>(arev, arep, W, prev, prep);
    bcast_relu_kernel<<<BATCH * SEQ * SEQ, HDIM / 4, 0, stream>>>(prev, prep, bias, (float*)d_out);
}